// DBLossWithShift_84817014162103
// MI455X (gfx1250) — compile-verified
//
#include <hip/hip_runtime.h>
#include <hip/hip_bf16.h>
#include <stdint.h>

#define B_      128
#define T_      8192
#define C_      8
#define NCHUNK  32
#define LCHUNK  (T_ / NCHUNK)      // 256 time steps per chunk
#define WARM    128                // EMA warm-up (0.8^128 ~ 4e-13)
#define TT      8                  // time steps per LDS tile
#define NTILES  (LCHUNK / TT)      // 32
#define BPB     32                 // b-rows per block
#define ROWF    72                 // floats per b-row in LDS (64 data + 8 pad, bank-conflict free)
#define ROWB    (ROWF * 4)         // 288 bytes
#define ARRF    (BPB * ROWF)       // 2304 floats per array per tile
#define ARRB    (ARRF * 4)         // 9216 bytes
#define BUFF    (2 * ARRF)         // both arrays
#define BUFB    (BUFF * 4)         // 18432 bytes
#define NBUF    3                  // triple buffer: 55296 B LDS

// ---- CDNA5 async global -> LDS copy (ASYNCcnt machinery) ----
__device__ __forceinline__ void async_cp16(uint32_t lds, const void* g) {
  asm volatile("global_load_async_to_lds_b128 %0, %1, off"
               :: "v"(lds), "v"(g) : "memory");
}
__device__ __forceinline__ void wait_async4() {
  asm volatile("s_wait_asynccnt 4" ::: "memory");
}
__device__ __forceinline__ void wait_async0() {
  asm volatile("s_wait_asynccnt 0" ::: "memory");
}

template <bool FIRST>
__device__ __forceinline__ void consume_tile(
    const float* __restrict__ sA, const float* __restrict__ sB,
    float& eh, float& eg, float (&rh)[8], float (&rg)[8],
    float& accdb, float (&accp)[6], float (&accn)[5]) {
#pragma unroll
  for (int u = 0; u < 8; ++u) {
    float xh = sA[u * 8];
    float xg = sB[u * 8];
    if (FIRST && u == 0) { eh = xh; eg = xg; }           // e[0] = x[0]
    else { eh = 0.2f * xh + 0.8f * eh; eg = 0.2f * xg + 0.8f * eg; }
    rh[u] = eh; rg[u] = eg;                              // ring idx == t & 7
    float dd = xh - xg;
    accdb += dd * dd;
#pragma unroll
    for (int d = 0; d < 6; ++d) {                        // shift +d: (eh[t]-e[t-d])^2
      if (!FIRST || u >= d) { float v = eh - rg[(u - d) & 7]; accp[d] += v * v; }
    }
#pragma unroll
    for (int s = 1; s < 6; ++s) {                        // shift -s: (eh[t-s]-e[t])^2
      if (!FIRST || u >= s) { float v = rh[(u - s) & 7] - eg; accn[s - 1] += v * v; }
    }
  }
}

__global__ __launch_bounds__(256)
void dbshift_main(const float* __restrict__ yh, const float* __restrict__ yg,
                  double* __restrict__ ws) {
  __shared__ float smem[NBUF * BUFF];

  const int tid   = threadIdx.x;
  const int blk   = blockIdx.x;
  const int chunk = blk >> 2;
  const int b0    = (blk & 3) * BPB;
  const int c     = tid & 7;
  const int bl    = tid >> 3;            // 0..31
  const int b     = b0 + bl;
  const int t0    = chunk * LCHUNK;

  const uint32_t lds_base = (uint32_t)(size_t)(void*)smem; // AS(3) offset = low 32 bits

  float eh = 0.f, eg = 0.f;
  float rh[8], rg[8];
#pragma unroll
  for (int i = 0; i < 8; ++i) { rh[i] = 0.f; rg[i] = 0.f; }
  float accdb = 0.f, accp[6], accn[5];
#pragma unroll
  for (int i = 0; i < 6; ++i) accp[i] = 0.f;
#pragma unroll
  for (int i = 0; i < 5; ++i) accn[i] = 0.f;

  const float* ph = yh + (size_t)b * T_ * C_ + c;
  const float* pg = yg + (size_t)b * T_ * C_ + c;

  // ---- warm-up: rebuild EMA state from t0-128 (carry error ~4e-13) ----
  if (chunk != 0) {
    const int ts = t0 - WARM;
#pragma unroll
    for (int u = 0; u < 8; ++u) {
      float xh = ph[(size_t)(ts + u) * C_];
      float xg = pg[(size_t)(ts + u) * C_];
      if (u == 0) { eh = xh; eg = xg; }
      else { eh = 0.2f * xh + 0.8f * eh; eg = 0.2f * xg + 0.8f * eg; }
      rh[u] = eh; rg[u] = eg;
    }
    for (int tb = ts + 8; tb < t0; tb += 8) {
#pragma unroll
      for (int u = 0; u < 8; ++u) {
        float xh = ph[(size_t)(tb + u) * C_];
        float xg = pg[(size_t)(tb + u) * C_];
        eh = 0.2f * xh + 0.8f * eh; eg = 0.2f * xg + 0.8f * eg;
        rh[u] = eh; rg[u] = eg;
      }
    }
  }

  // ---- triple-buffered async global->LDS pipeline over the owned chunk ----
  auto issue = [&](int tile, int buf) {
    const int tt = t0 + tile * TT;
#pragma unroll
    for (int i = 0; i < 2; ++i) {
      int lin = i * 256 + tid;           // 512 x 16B units per array
      int r   = lin >> 4;                // b-row 0..31 (256B contiguous run)
      int o   = (lin & 15) << 4;         // byte offset in run
      size_t goff = ((size_t)(b0 + r) * T_ + tt) * C_;
      uint32_t l0 = lds_base + (uint32_t)(buf * BUFB + r * ROWB + o);
      async_cp16(l0,        (const char*)(yh + goff) + o);
      async_cp16(l0 + ARRB, (const char*)(yg + goff) + o);
    }
  };

  issue(0, 0);
  issue(1, 1);
  for (int k = 0; k < NTILES; ++k) {
    if (k + 1 < NTILES) wait_async4();   // tile k done (k+1 may be in flight; in-order completion)
    else                wait_async0();
    __syncthreads();                     // tile k visible from all waves; buffer (k-1)%3 free
    if (k + 2 < NTILES) issue(k + 2, (k + 2) % NBUF);
    const int buf = k % NBUF;
    const float* sA = smem + buf * BUFF + bl * ROWF + c;
    const float* sB = sA + ARRF;
    if (chunk == 0 && k == 0)
      consume_tile<true >(sA, sB, eh, eg, rh, rg, accdb, accp, accn);
    else
      consume_tile<false>(sA, sB, eh, eg, rh, rg, accdb, accp, accn);
  }

  // ---- wave32 reduction, then f64 atomics (order-insensitive precision) ----
  float vals[12];
  vals[0] = accdb;
#pragma unroll
  for (int d = 0; d < 6; ++d) vals[1 + d] = accp[d];
#pragma unroll
  for (int s = 0; s < 5; ++s) vals[7 + s] = accn[s];
#pragma unroll
  for (int off = 16; off > 0; off >>= 1) {
#pragma unroll
    for (int i = 0; i < 12; ++i) vals[i] += __shfl_down(vals[i], off, 32);
  }
  if ((tid & 31) == 0) {
#pragma unroll
    for (int i = 0; i < 12; ++i) atomicAdd(&ws[i], (double)vals[i]);
  }
}

__global__ void dbshift_init(double* __restrict__ ws) {
  if (threadIdx.x < 12) ws[threadIdx.x] = 0.0;
}

__global__ void dbshift_finalize(const double* __restrict__ ws, float* __restrict__ out) {
  if (threadIdx.x == 0 && blockIdx.x == 0) {
    double db = ws[0] / (double)((size_t)B_ * T_ * C_);
    double m  = ws[1] / (1024.0 * 8192.0);                          // d = 0
    for (int d = 1; d < 6; ++d) {
      double v = ws[1 + d] / (1024.0 * (8192.0 - d));
      m = fmin(m, v);
    }
    for (int s = 1; s < 6; ++s) {
      double v = ws[6 + s] / (1024.0 * (8192.0 - s));
      m = fmin(m, v);
    }
    out[0] = (float)(db + 0.1 * m);
  }
}

extern "C" void kernel_launch(void* const* d_in, const int* in_sizes, int n_in,
                              void* d_out, int out_size, void* d_ws, size_t ws_size,
                              hipStream_t stream) {
  (void)in_sizes; (void)n_in; (void)out_size; (void)ws_size;
  const float* yh = (const float*)d_in[0];
  const float* yg = (const float*)d_in[1];
  double* ws  = (double*)d_ws;
  float*  out = (float*)d_out;
  hipLaunchKernelGGL(dbshift_init, dim3(1), dim3(32), 0, stream, ws);
  hipLaunchKernelGGL(dbshift_main, dim3(NCHUNK * 4), dim3(256), 0, stream, yh, yg, ws);
  hipLaunchKernelGGL(dbshift_finalize, dim3(1), dim3(1), 0, stream, ws, out);
}